// ConversationAwareRGCNLayer_19413252177999
// MI455X (gfx1250) — compile-verified
//
#include <hip/hip_runtime.h>
#include <hip/hip_bf16.h>

typedef __attribute__((ext_vector_type(2))) float v2f;
typedef __attribute__((ext_vector_type(8))) float v8f;

#define N_USER 100000
#define N_POST 100000
#define E_COM  1600000
#define E_UCU  1600000
#define IN_F   128
#define OUT_F  128
#define CONV_D 64
#define LN_EPS 1e-5f

static __device__ __forceinline__ v8f wmma_f32(v2f a, v2f b, v8f c) {
  // V_WMMA_F32_16X16X4_F32 : D = A(16x4) * B(4x16) + C(16x16), fp32
  return __builtin_amdgcn_wmma_f32_16x16x4_f32(
      /*neg_a=*/false, a, /*neg_b=*/false, b,
      /*c_mod=*/(short)0, c, /*reuse_a=*/false, /*reuse_b=*/false);
}

// ---------------------------------------------------------------------------
// Kernel 1: per-user transforms.
//   h_pub = h_user @ W_pub + b_pub
//   h_com = h_user @ W_com + b_com
//   z     = relu(LN(concat(h_user,user_ctx) @ W_conv + b_conv))
// One block = 16 user rows; 8 waves, wave w owns output cols [16w,16w+16).
// ---------------------------------------------------------------------------
__global__ void __launch_bounds__(256)
user_transform_kernel(const float* __restrict__ h_user,
                      const float* __restrict__ user_ctx,
                      const float* __restrict__ W_pub, const float* __restrict__ b_pub,
                      const float* __restrict__ W_com, const float* __restrict__ b_com,
                      const float* __restrict__ W_conv, const float* __restrict__ b_conv,
                      const float* __restrict__ ln_g, const float* __restrict__ ln_b,
                      float* __restrict__ hpub_out,
                      float* __restrict__ hcom_out,
                      float* __restrict__ z_out) {
  __shared__ float lds_hu[16 * IN_F];     // 8 KB
  __shared__ float lds_ctx[16 * CONV_D];  // 4 KB
  __shared__ float lds_z[16 * OUT_F];     // 8 KB
  __shared__ float lds_mu[16];
  __shared__ float lds_rs[16];

  const int t    = threadIdx.x;
  const int tile = blockIdx.x;          // 16 users per tile, N_USER % 16 == 0
  const int row0 = tile * 16;

  // Cooperative load of the A tiles (contiguous rows).
  {
    const float* src = h_user + (size_t)row0 * IN_F;
    for (int i = t; i < 16 * IN_F; i += 256) lds_hu[i] = src[i];
    const float* srcc = user_ctx + (size_t)row0 * CONV_D;
    for (int i = t; i < 16 * CONV_D; i += 256) lds_ctx[i] = srcc[i];
  }
  __syncthreads();

  const int lane   = t & 31;
  const int wave   = t >> 5;            // 0..7 -> output column slab
  const int m      = lane & 15;         // A row within tile
  const int khalf  = lane >> 4;         // 0: K={0,1}, 1: K={2,3}
  const int n      = wave * 16 + (lane & 15);  // output column

  v8f acc_pub = {};
  v8f acc_com = {};
  v8f acc_z   = {};

  // K = 0..127 over h_user
  for (int k0 = 0; k0 < IN_F; k0 += 4) {
    const int ka = k0 + 2 * khalf;
    v2f a  = { lds_hu[m * IN_F + ka], lds_hu[m * IN_F + ka + 1] };
    v2f bp = { W_pub[(size_t)ka * OUT_F + n], W_pub[(size_t)(ka + 1) * OUT_F + n] };
    v2f bc = { W_com[(size_t)ka * OUT_F + n], W_com[(size_t)(ka + 1) * OUT_F + n] };
    v2f bz = { W_conv[(size_t)ka * OUT_F + n], W_conv[(size_t)(ka + 1) * OUT_F + n] };
    acc_pub = wmma_f32(a, bp, acc_pub);
    acc_com = wmma_f32(a, bc, acc_com);
    acc_z   = wmma_f32(a, bz, acc_z);
  }
  // K = 128..191 over user_ctx (second half of W_conv)
  for (int k0 = 0; k0 < CONV_D; k0 += 4) {
    const int ka = k0 + 2 * khalf;
    v2f a  = { lds_ctx[m * CONV_D + ka], lds_ctx[m * CONV_D + ka + 1] };
    v2f bz = { W_conv[(size_t)(IN_F + ka) * OUT_F + n],
               W_conv[(size_t)(IN_F + ka + 1) * OUT_F + n] };
    acc_z = wmma_f32(a, bz, acc_z);
  }

  // Write pub/com results (+bias); stage raw z (+bias) in LDS for LayerNorm.
  const float bpn = b_pub[n];
  const float bcn = b_com[n];
  const float bzn = b_conv[n];
#pragma unroll
  for (int r = 0; r < 8; ++r) {
    const int mrow = r + 8 * khalf;              // C/D layout: M = r + 8*(lane>=16)
    const size_t grow = (size_t)(row0 + mrow);
    hpub_out[grow * OUT_F + n] = acc_pub[r] + bpn;
    hcom_out[grow * OUT_F + n] = acc_com[r] + bcn;
    lds_z[mrow * OUT_F + n]    = acc_z[r] + bzn;
  }
  __syncthreads();

  // Row-wise mean / rstd.
  if (t < 16) {
    float mu = 0.f;
    for (int c = 0; c < OUT_F; ++c) mu += lds_z[t * OUT_F + c];
    mu *= (1.0f / OUT_F);
    float var = 0.f;
    for (int c = 0; c < OUT_F; ++c) {
      float d = lds_z[t * OUT_F + c] - mu;
      var += d * d;
    }
    var *= (1.0f / OUT_F);
    lds_mu[t] = mu;
    lds_rs[t] = __frsqrt_rn(var + LN_EPS);
  }
  __syncthreads();

  // Apply LN * g + b, ReLU, store (contiguous 2048 floats per tile).
  for (int j = 0; j < 8; ++j) {
    const int i = t * 8 + j;
    const int r = i >> 7;
    const int c = i & 127;
    float v = (lds_z[i] - lds_mu[r]) * lds_rs[r] * ln_g[c] + ln_b[c];
    v = fmaxf(v, 0.0f);
    z_out[(size_t)row0 * OUT_F + i] = v;
  }
}

// ---------------------------------------------------------------------------
// Kernel 2: comment relation, fused e_proj GEMM + message + scatter-add.
//   msg = 0.7*h_com[src] + 0.3*(e_comment @ W_ecom + b_ecom)
// One block = 16 edges; 8 waves over output column slabs. E_COM % 16 == 0.
// ---------------------------------------------------------------------------
__global__ void __launch_bounds__(256)
ecom_scatter_kernel(const float* __restrict__ e_comment,
                    const int* __restrict__ com_src,
                    const int* __restrict__ com_dst,
                    const float* __restrict__ W_ecom, const float* __restrict__ b_ecom,
                    const float* __restrict__ hcom,
                    float* __restrict__ out_com,
                    float* __restrict__ cnt_com) {
  __shared__ float lds_ec[16 * CONV_D];   // 4 KB
  __shared__ int   lds_src[16];
  __shared__ int   lds_dst[16];

  const int t    = threadIdx.x;
  const int tile = blockIdx.x;
  const int e0   = tile * 16;

  {
    const float* src = e_comment + (size_t)e0 * CONV_D;
    for (int i = t; i < 16 * CONV_D; i += 256) lds_ec[i] = src[i];
  }
  if (t < 16) {
    lds_src[t] = com_src[e0 + t];
    lds_dst[t] = com_dst[e0 + t];
  }
  __syncthreads();

  const int lane  = t & 31;
  const int wave  = t >> 5;
  const int m     = lane & 15;
  const int khalf = lane >> 4;
  const int n     = wave * 16 + (lane & 15);

  v8f acc = {};
  for (int k0 = 0; k0 < CONV_D; k0 += 4) {
    const int ka = k0 + 2 * khalf;
    v2f a = { lds_ec[m * CONV_D + ka], lds_ec[m * CONV_D + ka + 1] };
    v2f b = { W_ecom[(size_t)ka * OUT_F + n], W_ecom[(size_t)(ka + 1) * OUT_F + n] };
    acc = wmma_f32(a, b, acc);
  }

  const float ben = b_ecom[n];
#pragma unroll
  for (int r = 0; r < 8; ++r) {
    const int mrow = r + 8 * khalf;
    const int src  = lds_src[mrow];
    const int dst  = lds_dst[mrow];
    const float msg = 0.7f * hcom[(size_t)src * OUT_F + n] + 0.3f * (acc[r] + ben);
    atomicAdd(&out_com[(size_t)dst * OUT_F + n], msg);
  }
  if (t < 16) atomicAdd(&cnt_com[lds_dst[t]], 1.0f);
}

// ---------------------------------------------------------------------------
// Generic row gather + scatter-add (pub and ucu relations).
// ---------------------------------------------------------------------------
__global__ void __launch_bounds__(256)
row_scatter_kernel(const float* __restrict__ vals,
                   const int* __restrict__ src,
                   const int* __restrict__ dst,
                   float* __restrict__ out,
                   float* __restrict__ cnt,
                   long long total /* n_edges * 128 */) {
  long long stride = (long long)gridDim.x * blockDim.x;
  for (long long i = (long long)blockIdx.x * blockDim.x + threadIdx.x; i < total; i += stride) {
    const int e = (int)(i >> 7);
    const int c = (int)(i & 127);
    const int d = dst[e];
    atomicAdd(&out[(size_t)d * OUT_F + c], vals[(size_t)src[e] * OUT_F + c]);
    if (c == 0) atomicAdd(&cnt[d], 1.0f);
  }
}

// ---------------------------------------------------------------------------
// Finalize: divide each segment sum by max(count, 1).
// ---------------------------------------------------------------------------
__global__ void __launch_bounds__(256)
finalize_kernel(float* __restrict__ out, const float* __restrict__ cnt, long long total) {
  long long stride = (long long)gridDim.x * blockDim.x;
  for (long long i = (long long)blockIdx.x * blockDim.x + threadIdx.x; i < total; i += stride) {
    const int row = (int)(i >> 7);               // flat row across the 3 stacked outputs
    out[i] /= fmaxf(cnt[row], 1.0f);
  }
}

// ---------------------------------------------------------------------------
extern "C" void kernel_launch(void* const* d_in, const int* in_sizes, int n_in,
                              void* d_out, int out_size, void* d_ws, size_t ws_size,
                              hipStream_t stream) {
  const float* h_user    = (const float*)d_in[0];
  // d_in[1] = h_post (unused by reference)
  const float* user_ctx  = (const float*)d_in[2];
  const float* e_comment = (const float*)d_in[3];
  const int*   pub_src   = (const int*)d_in[4];
  const int*   pub_dst   = (const int*)d_in[5];
  const int*   com_src   = (const int*)d_in[6];
  const int*   com_dst   = (const int*)d_in[7];
  const int*   ucu_src   = (const int*)d_in[8];
  const int*   ucu_dst   = (const int*)d_in[9];
  const float* W_pub     = (const float*)d_in[10];
  const float* b_pub     = (const float*)d_in[11];
  const float* W_com     = (const float*)d_in[12];
  const float* b_com     = (const float*)d_in[13];
  const float* W_conv    = (const float*)d_in[14];
  const float* b_conv    = (const float*)d_in[15];
  const float* ln_g      = (const float*)d_in[16];
  const float* ln_b      = (const float*)d_in[17];
  const float* W_ecom    = (const float*)d_in[18];
  const float* b_ecom    = (const float*)d_in[19];

  float* out      = (float*)d_out;
  float* out_pub  = out;                                     // [N_POST,128]
  float* out_com  = out + (size_t)N_POST * OUT_F;            // [N_POST,128]
  float* out_user = out + 2 * (size_t)N_POST * OUT_F;        // [N_USER,128]

  // Workspace layout (floats)
  float* hpub = (float*)d_ws;                                // 12.8M
  float* hcom = hpub + (size_t)N_USER * OUT_F;               // 12.8M
  float* zbuf = hcom + (size_t)N_USER * OUT_F;               // 12.8M
  float* cnt  = zbuf + (size_t)N_USER * OUT_F;               // 300K
  float* cnt_pub = cnt;
  float* cnt_com = cnt + N_POST;
  float* cnt_ucu = cnt + 2 * N_POST;

  // Zero accumulators (captured as memset nodes in the graph).
  hipMemsetAsync(d_out, 0, (size_t)3 * N_POST * OUT_F * sizeof(float), stream);
  hipMemsetAsync(cnt, 0, (size_t)3 * N_POST * sizeof(float), stream);

  // 1. User transforms (pub / com / conv+LN+ReLU), WMMA fp32.
  user_transform_kernel<<<N_USER / 16, 256, 0, stream>>>(
      h_user, user_ctx, W_pub, b_pub, W_com, b_com, W_conv, b_conv,
      ln_g, ln_b, hpub, hcom, zbuf);

  // 2. Comment relation: fused e_proj GEMM + message + scatter.
  ecom_scatter_kernel<<<E_COM / 16, 256, 0, stream>>>(
      e_comment, com_src, com_dst, W_ecom, b_ecom, hcom, out_com, cnt_com);

  // 3. Publish relation scatter (100K edges).
  {
    long long total = (long long)100000 * OUT_F;
    int blocks = (int)((total + 255) / 256);
    row_scatter_kernel<<<blocks, 256, 0, stream>>>(hpub, pub_src, pub_dst,
                                                   out_pub, cnt_pub, total);
  }
  // 4. user-comment-user relation scatter (1.6M edges).
  {
    long long total = (long long)E_UCU * OUT_F;
    int blocks = (int)((total + 255) / 256);
    row_scatter_kernel<<<blocks, 256, 0, stream>>>(zbuf, ucu_src, ucu_dst,
                                                   out_user, cnt_ucu, total);
  }
  // 5. Divide sums by counts -> means.
  {
    long long total = (long long)3 * N_POST * OUT_F;
    int blocks = (int)((total + 255) / 256);
    finalize_kernel<<<blocks, 256, 0, stream>>>(out, cnt, total);
  }
}